// Encoder_56384330662538
// MI455X (gfx1250) — compile-verified
//
#include <hip/hip_runtime.h>
#include <math.h>

typedef __attribute__((ext_vector_type(2))) float v2f;
typedef __attribute__((ext_vector_type(8))) float v8f;

#define VOCAB  32000
#define EMBED  256
#define HIDDEN 512
#define BATCH  64
#define SEQ    512

#define NWG         8                    // workgroups in recurrent kernel
#define COLS_PER_WG (HIDDEN / NWG)       // 64 columns of W_hh per WG
#define KPAD        (HIDDEN + 4)         // 516: 16B-aligned float4 rows,
                                         // conflict-free b64 fragment loads

// ---------------------------------------------------------------------------
// Kernel 1 (fully parallel): X[m,n] = sum_k emb[src[m],k]*W_ih[k,n] + b_ih[n]
//                            + b_hh[n], m in [0, B*S), n in [0, H).
// One wave per 16x16 output tile, f32 WMMA (16x16x4), K = 256 -> 64 wmma ops
// on two interleaved accumulators. Written straight into d_out; consumed
// (and overwritten) by the RNN kernel.
// ---------------------------------------------------------------------------
__global__ void __launch_bounds__(256)
proj_kernel(const int* __restrict__ src, const float* __restrict__ emb,
            const float* __restrict__ Wih, const float* __restrict__ bih,
            const float* __restrict__ bhh, float* __restrict__ out)
{
    const int lane = threadIdx.x & 31;
    const int wave = threadIdx.x >> 5;
    const int half = lane >> 4;          // 0: lanes 0-15, 1: lanes 16-31
    const int l16  = lane & 15;
    const int m0   = blockIdx.x * 128 + wave * 16;   // 8 waves cover 128 rows
    const int n0   = blockIdx.y * 16;

    // A-fragment source row: gathered embedding row for token at flat index m.
    const int m     = m0 + l16;
    const int token = src[m];
    const float* arow = emb + (size_t)token * EMBED + 2 * half;
    // B-fragment source: W_ih rows k+2h, k+2h+1 at column n0+l16.
    const float* bcol = Wih + (size_t)(2 * half) * HIDDEN + n0 + l16;

    // c0 init = fused bias (same N per lane for all 8 accumulator rows).
    const float bias = bih[n0 + l16] + bhh[n0 + l16];
    v8f c0, c1;
#pragma unroll
    for (int r = 0; r < 8; ++r) { c0[r] = bias; c1[r] = 0.0f; }

#pragma unroll 4
    for (int k = 0; k < EMBED; k += 8) {
        v2f a0 = *(const v2f*)(arow + k);
        v2f a1 = *(const v2f*)(arow + k + 4);
        v2f b0, b1;
        b0.x = bcol[(size_t)k * HIDDEN];
        b0.y = bcol[(size_t)(k + 1) * HIDDEN];
        b1.x = bcol[(size_t)(k + 4) * HIDDEN];
        b1.y = bcol[(size_t)(k + 5) * HIDDEN];
        c0 = __builtin_amdgcn_wmma_f32_16x16x4_f32(false, a0, false, b0,
                                                   (short)0, c0, false, false);
        c1 = __builtin_amdgcn_wmma_f32_16x16x4_f32(false, a1, false, b1,
                                                   (short)0, c1, false, false);
    }

    // C layout: VGPR r -> row r (lanes 0-15) / row r+8 (lanes 16-31).
    float* orow = out + (size_t)(m0 + 8 * half) * HIDDEN + n0 + l16;
#pragma unroll
    for (int r = 0; r < 8; ++r) orow[(size_t)r * HIDDEN] = c0[r] + c1[r];
}

// ---------------------------------------------------------------------------
// Kernel 2 (persistent, sequential over t): h_t = tanh(X_t + h_{t-1} W_hh).
// 8 WGs x 512 threads (16 waves). Each WG owns 64 columns of W_hh, stored
// TRANSPOSED in LDS once. Per step: stage full h_{t-1} from d_out with
// coalesced b128 loads, 16 waves each do one 16x16 tile of the 64x64 output
// slice via 2x64 interleaved f32 WMMAs, tanh, store to d_out, then a
// monotonic atomic global barrier (release fence before, acquire fence by
// every thread after) releases step t+1.
// ---------------------------------------------------------------------------
__global__ void __launch_bounds__(512)
rnn_kernel(const float* __restrict__ Whh, float* out, unsigned int* counter)
{
    extern __shared__ float smem[];
    float* WT = smem;                          // [COLS_PER_WG][KPAD] (W^T slice)
    float* hs = smem + COLS_PER_WG * KPAD;     // [BATCH][KPAD]      (h_{t-1})

    const int tid   = threadIdx.x;
    const int lane  = tid & 31;
    const int wave  = tid >> 5;
    const int half  = lane >> 4;
    const int l16   = lane & 15;
    const int nbase = blockIdx.x * COLS_PER_WG;

    // Load resident W_hh slice, transposed: WT[c][k] = W_hh[k][nbase+c].
    {
        const int c = tid & (COLS_PER_WG - 1);     // tid % 64  (coalesced k-rows)
        for (int k = (tid >> 6); k < HIDDEN; k += 8)
            WT[c * KPAD + k] = Whh[(size_t)k * HIDDEN + nbase + c];
    }
    __syncthreads();

    const int mi = wave & 3;                       // M tile (batch rows)
    const int ni = wave >> 2;                      // N tile (hidden cols)
    const int n_local = ni * 16 + l16;
    const int n       = nbase + n_local;

    // h staging map: thread owns float4 column group c4, rows b0, b0+4, ...
    const int c4 = (tid & 127) * 4;                // 128 groups x 4 cols
    const int b0 = tid >> 7;                       // 0..3

    for (int t = 0; t < SEQ; ++t) {
        // Accumulator 0 init = X fragment (this WG's columns at time t).
        v8f c0, c1;
        {
            const size_t base = (size_t)(mi * 16 + 8 * half) * SEQ * HIDDEN
                              + (size_t)t * HIDDEN + n;
#pragma unroll
            for (int r = 0; r < 8; ++r) {
                c0[r] = out[base + (size_t)r * SEQ * HIDDEN];
                c1[r] = 0.0f;
            }
        }

        if (t > 0) {
            // Stage h_{t-1} = d_out[:, t-1, :] into LDS (b128, coalesced).
#pragma unroll 4
            for (int b = b0; b < BATCH; b += 4) {
                float4 v = *(const float4*)(out + (size_t)b * SEQ * HIDDEN
                                                + (size_t)(t - 1) * HIDDEN + c4);
                *(float4*)(hs + b * KPAD + c4) = v;
            }
            __syncthreads();

            const float* arow = hs + (size_t)(mi * 16 + l16) * KPAD + 2 * half;
            const float* brow = WT + (size_t)n_local * KPAD + 2 * half;
#pragma unroll 8
            for (int k = 0; k < HIDDEN; k += 8) {
                v2f a0 = *(const v2f*)(arow + k);       // h[m][k+2h .. +1]
                v2f b0v = *(const v2f*)(brow + k);      // W[k+2h .. +1][n]
                v2f a1 = *(const v2f*)(arow + k + 4);
                v2f b1v = *(const v2f*)(brow + k + 4);
                c0 = __builtin_amdgcn_wmma_f32_16x16x4_f32(false, a0, false, b0v,
                                                           (short)0, c0, false, false);
                c1 = __builtin_amdgcn_wmma_f32_16x16x4_f32(false, a1, false, b1v,
                                                           (short)0, c1, false, false);
            }
            __syncthreads();                       // hs reused next step
        }

        // h_t = tanh(acc); write back over X_t (also the final output).
        {
            const size_t base = (size_t)(mi * 16 + 8 * half) * SEQ * HIDDEN
                              + (size_t)t * HIDDEN + n;
#pragma unroll
            for (int r = 0; r < 8; ++r)
                out[base + (size_t)r * SEQ * HIDDEN] = tanhf(c0[r] + c1[r]);
        }

        // Device-wide barrier. Release: every thread flushes its stores.
        __threadfence();
        __syncthreads();
        if (tid == 0) {
            __hip_atomic_fetch_add(counter, 1u, __ATOMIC_RELEASE,
                                   __HIP_MEMORY_SCOPE_AGENT);
            const unsigned want = (unsigned)(t + 1) * NWG;
            while (__hip_atomic_load(counter, __ATOMIC_ACQUIRE,
                                     __HIP_MEMORY_SCOPE_AGENT) < want)
                __builtin_amdgcn_s_sleep(1);
        }
        __syncthreads();
        // Acquire: every thread (all waves / both CUs of the WGP) invalidates
        // stale cached lines before reading other WGs' h_t next iteration.
        __threadfence();
    }
}

// Reset barrier counter every launch (graph replays reuse d_ws).
__global__ void init_counter(unsigned int* c)
{
    if (threadIdx.x == 0) *c = 0u;
}

extern "C" void kernel_launch(void* const* d_in, const int* in_sizes, int n_in,
                              void* d_out, int out_size, void* d_ws, size_t ws_size,
                              hipStream_t stream)
{
    (void)in_sizes; (void)n_in; (void)out_size; (void)ws_size;
    const int*   src = (const int*)d_in[0];
    const float* emb = (const float*)d_in[1];
    const float* Wih = (const float*)d_in[2];
    const float* Whh = (const float*)d_in[3];
    const float* bih = (const float*)d_in[4];
    const float* bhh = (const float*)d_in[5];
    float*       out = (float*)d_out;
    unsigned int* counter = (unsigned int*)d_ws;

    init_counter<<<1, 64, 0, stream>>>(counter);

    dim3 g1((BATCH * SEQ) / 128, HIDDEN / 16);     // 256 x 32 blocks
    proj_kernel<<<g1, 256, 0, stream>>>(src, emb, Wih, bih, bhh, out);

    const size_t lds_bytes = (size_t)(COLS_PER_WG + BATCH) * KPAD * sizeof(float);
    rnn_kernel<<<NWG, 512, lds_bytes, stream>>>(Whh, out, counter);
}